// SPDUnVectorize_13546326851714
// MI455X (gfx1250) — compile-verified
//
#include <hip/hip_runtime.h>
#include <stdint.h>

#define N_MAT   256
#define D_PACK  32896          // N*(N+1)/2
#define T       64             // tile edge
#define LDSS    65             // padded LDS row stride (65 % 64 == 1 -> conflict-free cols)
#define NPAIR   10             // upper-triangular 4x4 tile pairs

#if __has_builtin(__builtin_amdgcn_global_load_async_to_lds_b32)
#define HAVE_ASYNC_LD 1
#else
#define HAVE_ASYNC_LD 0
#endif

#if __has_builtin(__builtin_amdgcn_global_store_async_from_lds_b32)
#define HAVE_ASYNC_ST 1
#else
#define HAVE_ASYNC_ST 0
#endif

// Builtin parameter types (from clang diagnostics): arg0 = AS(1) int*, arg1 = AS(3) int*
typedef __attribute__((address_space(1))) int GInt;
typedef __attribute__((address_space(3))) int LInt;

__device__ __forceinline__ int row_start(int i) {
    // packed index of first element of row i of the upper triangle
    return i * N_MAT - (i * (i - 1)) / 2;
}

__global__ __launch_bounds__(256) void spd_unvec_kernel(const float* __restrict__ x,
                                                        float* __restrict__ out) {
    __shared__ float tile[T * LDSS];

    const int pair = blockIdx.x % NPAIR;
    const int b    = blockIdx.x / NPAIR;

    // decode upper-triangular tile pair (ti <= tj) of the 4x4 tile grid
    int ti = 0, rem = pair;
    while (rem >= (4 - ti)) { rem -= (4 - ti); ++ti; }
    const int tj = ti + rem;
    const int i0 = ti * T, j0 = tj * T;

    const float* xb = x + (size_t)b * D_PACK;
    float*       ob = out + (size_t)b * (N_MAT * N_MAT);

    const int tid = threadIdx.x;
    const int lc  = tid & 63;   // col within tile
    const int lr4 = tid >> 6;   // 0..3 row sub-index

    // ---- stage tile rows [i0, i0+64), cols [j0, j0+64) from packed storage into LDS ----
    // Row i of the tile is the contiguous packed run starting at rowStart(i) + j0 - i.
    // (Diagonal tiles: lanes with j < i read harmless in-bounds garbage, never consumed.)
#pragma unroll
    for (int p = 0; p < 16; ++p) {
        const int r   = p * 4 + lr4;
        const int i   = i0 + r;
        const int src = row_start(i) + (j0 + lc) - i;
#if HAVE_ASYNC_LD
        __builtin_amdgcn_global_load_async_to_lds_b32(
            (GInt*)(xb + src),
            (LInt*)&tile[r * LDSS + lc],
            0, 0);
#else
        tile[r * LDSS + lc] = xb[src];
#endif
    }

#if HAVE_ASYNC_LD
#if __has_builtin(__builtin_amdgcn_s_wait_asynccnt)
    __builtin_amdgcn_s_wait_asynccnt(0);
#else
    asm volatile("s_wait_asynccnt 0" ::: "memory");
#endif
#endif
    __syncthreads();

    // ---- emit: direct image at (i0,j0); transposed image at (j0,i0) for off-diag tiles ----
    // All global stores are wave-contiguous (coalesced). LDS reads:
    //   direct:     tile[r*65 + c]  (stride-1 across lanes, conflict-free)
    //   transposed: tile[c*65 + r]  (stride-65 across lanes, 65%64==1 -> conflict-free)
#pragma unroll
    for (int p = 0; p < 16; ++p) {
        const int r = p * 4 + lr4;
        const int c = lc;

        // direct image; diagonal tiles symmetrize purely via LDS-address select
        const int lsrc0 = (ti == tj && r > c) ? (c * LDSS + r) : (r * LDSS + c);
        float* g0 = ob + (size_t)(i0 + r) * N_MAT + (j0 + c);
#if HAVE_ASYNC_ST
        __builtin_amdgcn_global_store_async_from_lds_b32(
            (GInt*)g0, (LInt*)&tile[lsrc0], 0, 0);
#else
        *g0 = tile[lsrc0];
#endif

        if (ti != tj) {
            // transposed image: out[j0+r][i0+c] = in(i0+c, j0+r) = tile[c][r]
            const int lsrc1 = c * LDSS + r;
            float* g1 = ob + (size_t)(j0 + r) * N_MAT + (i0 + c);
#if HAVE_ASYNC_ST
            __builtin_amdgcn_global_store_async_from_lds_b32(
                (GInt*)g1, (LInt*)&tile[lsrc1], 0, 0);
#else
            *g1 = tile[lsrc1];
#endif
        }
    }
    // s_endpgm performs an implicit wait-idle, covering outstanding async stores.
}

extern "C" void kernel_launch(void* const* d_in, const int* in_sizes, int n_in,
                              void* d_out, int out_size, void* d_ws, size_t ws_size,
                              hipStream_t stream) {
    (void)n_in; (void)d_ws; (void)ws_size; (void)out_size;
    const float* x   = (const float*)d_in[0];
    float*       out = (float*)d_out;
    const int B = in_sizes[0] / D_PACK;   // B * N(N+1)/2 flat elements
    dim3 grid(B * NPAIR);
    dim3 block(256);
    spd_unvec_kernel<<<grid, block, 0, stream>>>(x, out);
}